// VPLayer_tf_24275155157103
// MI455X (gfx1250) — compile-verified
//
#include <hip/hip_runtime.h>
#include <math.h>

typedef __attribute__((ext_vector_type(2))) float v2f;
typedef __attribute__((ext_vector_type(8))) float v8f;

#define N_IN  512
#define N_VP  12
#define NPAD  16
#define BATCH 65536

// ---------------------------------------------------------------------------
// Kernel 1 (tiny, one block): A = phi * (phi^T phi)^-1, padded to [512 x 16].
// pinv(phi) = (phi^T phi)^-1 phi^T  =>  out = x @ phi @ Ginv = x @ A.
// ---------------------------------------------------------------------------
__global__ __launch_bounds__(512) void build_proj_kernel(
    const float* __restrict__ w, float* __restrict__ Apad) {
  __shared__ float sphi[N_IN][N_VP];
  __shared__ float sG[N_VP][N_VP];
  __shared__ float sInv[N_VP][N_VP];
  const int tid = threadIdx.x;
  const float dil = w[0], tr = w[1];

  // Weighted Hermite recurrence for this sample.
  {
    const float t  = (float)tid - 0.5f * (float)(N_IN - 1);
    const float xs = dil * (t - tr);
    const float pi14 = 0.75112554446494248f;  // pi^-0.25
    float pm2 = pi14 * __expf(-0.5f * xs * xs);
    sphi[tid][0] = pm2;
    float pm1 = 1.41421356237309515f * xs * pm2;
    sphi[tid][1] = pm1;
    for (int k = 2; k < N_VP; ++k) {
      float p = sqrtf(2.0f / (float)k) * xs * pm1
              - sqrtf((float)(k - 1) / (float)k) * pm2;
      sphi[tid][k] = p;
      pm2 = pm1; pm1 = p;
    }
  }
  __syncthreads();

  // Gram matrix G = phi^T phi (12x12), one entry per thread.
  if (tid < N_VP * N_VP) {
    const int j = tid / N_VP, k = tid % N_VP;
    float s = 0.0f;
    for (int i = 0; i < N_IN; ++i) s += sphi[i][j] * sphi[i][k];
    sG[j][k] = s;
  }
  __syncthreads();

  // 12x12 Gauss-Jordan inverse with partial pivoting (single thread; trivial).
  if (tid == 0) {
    for (int r = 0; r < N_VP; ++r)
      for (int c2 = 0; c2 < N_VP; ++c2)
        sInv[r][c2] = (r == c2) ? 1.0f : 0.0f;
    for (int col = 0; col < N_VP; ++col) {
      int piv = col;
      float best = fabsf(sG[col][col]);
      for (int r = col + 1; r < N_VP; ++r) {
        float v = fabsf(sG[r][col]);
        if (v > best) { best = v; piv = r; }
      }
      if (piv != col) {
        for (int c2 = 0; c2 < N_VP; ++c2) {
          float t0 = sG[col][c2];  sG[col][c2]  = sG[piv][c2];  sG[piv][c2]  = t0;
          float t1 = sInv[col][c2]; sInv[col][c2] = sInv[piv][c2]; sInv[piv][c2] = t1;
        }
      }
      const float inv = 1.0f / sG[col][col];
      for (int c2 = 0; c2 < N_VP; ++c2) { sG[col][c2] *= inv; sInv[col][c2] *= inv; }
      for (int r = 0; r < N_VP; ++r) {
        if (r == col) continue;
        const float f = sG[r][col];
        for (int c2 = 0; c2 < N_VP; ++c2) {
          sG[r][c2]  -= f * sG[col][c2];
          sInv[r][c2] -= f * sInv[col][c2];
        }
      }
    }
  }
  __syncthreads();

  // A[k][col] = sum_j phi[k][j] * Ginv[j][col], zero-padded cols 12..15.
  for (int col = 0; col < NPAD; ++col) {
    float s = 0.0f;
    if (col < N_VP)
      for (int j = 0; j < N_VP; ++j) s += sphi[tid][j] * sInv[j][col];
    Apad[tid * NPAD + col] = s;
  }
}

// ---------------------------------------------------------------------------
// Kernel 2: out[65536,12] = x[65536,512] @ Apad[512,16]  via f32 WMMA.
// 4 waves/block; each wave owns a 16-row tile, K-loop of 128 x wmma 16x16x4.
// A-frag layout (ISA 7.12.2, 32-bit A 16x4): lane(m,hi) holds x[m][k+2hi+{0,1}]
//   -> one contiguous 8B global load per lane per step.
// B-frag: lane(n,hi) holds Apad[k+2hi][n], Apad[k+1+2hi][n] from LDS
//   -> conflict-free ds_load_b32 (16 consecutive banks per half-wave).
// ---------------------------------------------------------------------------
__global__ __launch_bounds__(128) void coeff_gemm_kernel(
    const float* __restrict__ x, const float* __restrict__ Apad,
    float* __restrict__ out) {
  __shared__ float sB[N_IN * NPAD];  // 32 KB, resident per block
  for (int idx = threadIdx.x; idx < N_IN * NPAD; idx += 128)
    sB[idx] = Apad[idx];
  __syncthreads();

  const int lane = threadIdx.x & 31;
  const int wave = threadIdx.x >> 5;
  const int n    = lane & 15;   // A-frag row m == B-frag col n == lane&15
  const int hi   = lane >> 4;
  const int row_base = (blockIdx.x * 4 + wave) * 16;

  const float* xp = x + (size_t)(row_base + n) * N_IN + 2 * hi;
  const float* bp = sB + (2 * hi) * NPAD + n;

  v8f c = {0.f, 0.f, 0.f, 0.f, 0.f, 0.f, 0.f, 0.f};
#pragma unroll 4
  for (int k = 0; k < N_IN; k += 4) {
    v2f a = *(const v2f*)(xp + k);          // global_load_b64, streams x once
    v2f b;
    b.x = bp[(k + 0) * NPAD];               // ds_load_b32
    b.y = bp[(k + 1) * NPAD];               // ds_load_b32
    // D = A(16x4,f32) x B(4x16,f32) + C(16x16,f32)
    c = __builtin_amdgcn_wmma_f32_16x16x4_f32(
        /*neg_a=*/false, a, /*neg_b=*/false, b,
        /*c_mod=*/(short)0, c, /*reuse_a=*/false, /*reuse_b=*/false);
  }

  if (n < N_VP) {
    float* op = out + (size_t)(row_base + 8 * hi) * N_VP + n;
#pragma unroll
    for (int i = 0; i < 8; ++i)
      op[i * N_VP] = c[i];                  // D: vgpr i -> row (i + 8*hi), col n
  }
}

// ---------------------------------------------------------------------------
extern "C" void kernel_launch(void* const* d_in, const int* in_sizes, int n_in,
                              void* d_out, int out_size, void* d_ws, size_t ws_size,
                              hipStream_t stream) {
  (void)in_sizes; (void)n_in; (void)out_size; (void)ws_size;
  const float* x = (const float*)d_in[0];   // [65536, 512]
  const float* w = (const float*)d_in[1];   // [2]
  float* out  = (float*)d_out;              // [65536, 12]
  float* Apad = (float*)d_ws;               // [512, 16] scratch (32 KB)

  build_proj_kernel<<<1, 512, 0, stream>>>(w, Apad);
  coeff_gemm_kernel<<<BATCH / 64, 128, 0, stream>>>(x, Apad, out);
}